// GAT_85014582657621
// MI455X (gfx1250) — compile-verified
//
#include <hip/hip_runtime.h>

// ---------------------------------------------------------------------------
// GAT-style attention aggregate, MI455X (gfx1250, wave32, WMMA).
//   B=8, S=512, N=32, H=256, V=100001.  Memory-bound gather (~135 MB; table
//   fits in 192MB L2).  f32 end-to-end; w^T x Cand done on v_wmma_f32_16x16x4_f32.
// ---------------------------------------------------------------------------

constexpr int kB = 8;
constexpr int kS = 512;
constexpr int kN = 32;      // neighbors
constexpr int kH = 256;     // hidden
constexpr int kRows = kN + 1;     // 33 candidates (self + neighbors)
constexpr int kRowsPad = 36;      // padded to multiple of 4 (WMMA K-chunks)
constexpr float kNeg = -1e9f;
constexpr float kSlope = 0.2f;    // LeakyReLU

typedef __attribute__((ext_vector_type(2))) float v2f;
typedef __attribute__((ext_vector_type(8))) float v8f;

__global__ __launch_bounds__(256) void gat_kernel(
    const int* __restrict__ node_ids,    // [B*S]
    const int* __restrict__ neighs,      // [B*S*N]
    const int* __restrict__ mask,        // [B*S*N]
    const float* __restrict__ emb,       // [V*H]
    const float* __restrict__ a_w,       // [2H]
    const float* __restrict__ a_b,       // [1]
    float* __restrict__ out)             // [B*S*H]
{
    __shared__ float sh_cand[kRowsPad][kH];   // 36 KB candidate tile
    __shared__ float sh_z[kRows];             // per-row dot(cand, a2), then att
    __shared__ float sh_w[kRowsPad];          // softmax weights (pad = 0)
    __shared__ float sh_dot1;                 // dot(src, a1)

    const int pos  = blockIdx.x;              // 0 .. B*S-1
    const int tid  = threadIdx.x;
    const int wave = tid >> 5;                // 0..7 (wave32)
    const int lane = tid & 31;

    // Per-lane 8-float slice of a2 = a_w[H:2H], reused across all owned rows.
    float a2r[8];
    #pragma unroll
    for (int i = 0; i < 8; ++i) a2r[i] = a_w[kH + lane * 8 + i];

    // ---- zero the 3 pad rows (rows 33..35): waves 1..3 -------------------
    if (wave >= 1 && wave <= 3) {
        float4 z4 = make_float4(0.f, 0.f, 0.f, 0.f);
        float4* sp = (float4*)(&sh_cand[32 + wave][lane * 8]);
        sp[0] = z4; sp[1] = z4;
    }

    // ---- row 0 (self): wave 0 loads, stores to LDS, dots with a1 and a2 --
    if (wave == 0) {
        const int idx = node_ids[pos];
        const float4* rp = (const float4*)(emb + (size_t)idx * kH) + lane * 2;
        float4 r0 = rp[0];
        float4 r1 = rp[1];
        float4* sp = (float4*)(&sh_cand[0][lane * 8]);
        sp[0] = r0; sp[1] = r1;

        float p2 = r0.x*a2r[0] + r0.y*a2r[1] + r0.z*a2r[2] + r0.w*a2r[3]
                 + r1.x*a2r[4] + r1.y*a2r[5] + r1.z*a2r[6] + r1.w*a2r[7];
        const float* a1 = a_w + lane * 8;
        float p1 = r0.x*a1[0] + r0.y*a1[1] + r0.z*a1[2] + r0.w*a1[3]
                 + r1.x*a1[4] + r1.y*a1[5] + r1.z*a1[6] + r1.w*a1[7];
        #pragma unroll
        for (int off = 16; off > 0; off >>= 1) {
            p2 += __shfl_xor(p2, off, 32);
            p1 += __shfl_xor(p1, off, 32);
        }
        if (lane == 0) { sh_z[0] = p2; sh_dot1 = p1; }
    }

    // ---- neighbor rows: 32 rows striped over 8 waves (4 each) ------------
    for (int jj = wave; jj < kN; jj += 8) {
        const int idx = neighs[pos * kN + jj];
        // prefetch next owned neighbor row while this one is in flight
        if (jj + 8 < kN) {
            const int nidx = neighs[pos * kN + jj + 8];
            __builtin_prefetch(emb + (size_t)nidx * kH + lane * 8, 0, 1);
        }
        const float4* rp = (const float4*)(emb + (size_t)idx * kH) + lane * 2;
        float4 r0 = rp[0];
        float4 r1 = rp[1];
        float4* sp = (float4*)(&sh_cand[jj + 1][lane * 8]);
        sp[0] = r0; sp[1] = r1;

        float p2 = r0.x*a2r[0] + r0.y*a2r[1] + r0.z*a2r[2] + r0.w*a2r[3]
                 + r1.x*a2r[4] + r1.y*a2r[5] + r1.z*a2r[6] + r1.w*a2r[7];
        #pragma unroll
        for (int off = 16; off > 0; off >>= 1)
            p2 += __shfl_xor(p2, off, 32);
        if (lane == 0) sh_z[jj + 1] = p2;
    }

    __syncthreads();

    // ---- logits: z = dot1 + dot2 + b -> LeakyReLU -> neighbor mask -------
    if (tid < kRows) {
        float z = sh_dot1 + sh_z[tid] + a_b[0];
        float att = z > 0.f ? z : kSlope * z;
        if (tid >= 1) att += (float)mask[pos * kN + (tid - 1)] * kNeg;
        sh_z[tid] = att;
    }
    __syncthreads();

    // ---- softmax over 33 scalars (serial; trivial vs. memory time) -------
    if (tid == 0) {
        float m = sh_z[0];
        for (int j = 1; j < kRows; ++j) m = fmaxf(m, sh_z[j]);
        float s = 0.f;
        for (int j = 0; j < kRows; ++j) s += __expf(sh_z[j] - m);
        float inv = 1.f / s;
        for (int j = 0; j < kRows; ++j) sh_w[j] = __expf(sh_z[j] - m) * inv;
        sh_w[33] = 0.f; sh_w[34] = 0.f; sh_w[35] = 0.f;
    }
    __syncthreads();

    // ---- aggregation: out[1,256] = w[1,36] x Cand[36,256] via WMMA -------
    // V_WMMA_F32_16X16X4_F32:
    //   A (16x4, all rows = same w chunk): lanes 0-15 -> K0,K1; lanes 16-31 -> K2,K3
    //   B (4x16): lane&15 = column; lanes 0-15 -> K0,K1; lanes 16-31 -> K2,K3
    //   D (16x16): VGPR0, lanes 0-15 = row M=0  -> our output tile.
    // Loop trip count is uniform across the wave (16 tiles / 8 waves = 2), so
    // EXEC is all-1s at every WMMA as required.
    const int kgrp = (lane >> 4) << 1;   // 0 or 2
    const int col  = lane & 15;
    for (int c = wave; c < kH / 16; c += 8) {
        v8f acc = {};
        #pragma unroll
        for (int k0 = 0; k0 < kRowsPad; k0 += 4) {
            v2f a, bm;
            a.x  = sh_w[k0 + kgrp + 0];
            a.y  = sh_w[k0 + kgrp + 1];
            bm.x = sh_cand[k0 + kgrp + 0][c * 16 + col];
            bm.y = sh_cand[k0 + kgrp + 1][c * 16 + col];
            acc = __builtin_amdgcn_wmma_f32_16x16x4_f32(
                /*neg_a=*/false, a, /*neg_b=*/false, bm,
                /*c_mod=*/(short)0, acc, /*reuse_a=*/false, /*reuse_b=*/false);
        }
        if (lane < 16)
            out[(size_t)pos * kH + c * 16 + lane] = acc[0];
    }
}

extern "C" void kernel_launch(void* const* d_in, const int* in_sizes, int n_in,
                              void* d_out, int out_size, void* d_ws, size_t ws_size,
                              hipStream_t stream) {
    const int*   node_ids = (const int*)d_in[0];
    const int*   neighs   = (const int*)d_in[1];
    const int*   mask     = (const int*)d_in[2];
    const float* emb      = (const float*)d_in[3];
    const float* a_w      = (const float*)d_in[4];
    const float* a_b      = (const float*)d_in[5];
    float* out = (float*)d_out;

    dim3 grid(kB * kS);   // 4096 workgroups, one per (b,s) position
    dim3 block(256);      // 8 wave32 waves
    gat_kernel<<<grid, block, 0, stream>>>(node_ids, neighs, mask, emb, a_w, a_b, out);
}